// SetConv_8469675508025
// MI455X (gfx1250) — compile-verified
//
#include <hip/hip_runtime.h>

// Problem constants (from setup_inputs): fixed sizes, hardcode for codegen.
#define NC      1024      // context points
#define NG      2048      // grid points
#define OUTD    32        // out_dim
#define CH      33        // in_chan = 1 + 32
#define KPAD    36        // CH padded to multiple of 4 for WMMA K-steps
#define NSLICE  8         // i-dimension split for parallelism (deterministic partials)
#define ISLICE  (NC / NSLICE)   // 128
#define TJ      256       // threads per block in accumulation kernel
#define LOG2E   1.4426950408889634f

typedef float v2f __attribute__((ext_vector_type(2)));
typedef float v8f __attribute__((ext_vector_type(8)));

// ---------------------------------------------------------------------------
// Kernel 0: pad W (32x33) into Wp (32x36, zero-filled cols 33..35) so the
// WMMA B-fragment loads are unconditional contiguous b64s (no EXEC-masked
// ternary loads in the hot GEMM wave).
// ---------------------------------------------------------------------------
__global__ __launch_bounds__(256) void prep_w_kernel(
    const float* __restrict__ W, float* __restrict__ Wp)
{
    int idx = blockIdx.x * blockDim.x + threadIdx.x;   // 0 .. 32*36-1
    if (idx >= OUTD * KPAD) return;
    int o = idx / KPAD, c = idx % KPAD;
    Wp[idx] = (c < CH) ? W[o * CH + c] : 0.0f;
}

// ---------------------------------------------------------------------------
// Kernel 1: partial set-conv accumulation.
//   part[s][j][c] = sum_{i in slice s} Fc[i,c] * exp2(d(i,j) * gamma_c)
// gamma_c = -0.5 * exp(-2*log_sigma[c]) * log2(e)  (one v_exp_f32 per term)
// One thread per grid point j; i-slices across blockIdx.y. Context slice staged
// in LDS; gamma + 33 accumulators live in VGPRs. Transcendental-bound by design
// (69M v_exp_f32 total — the intrinsic cost of per-channel kernel weights).
// ---------------------------------------------------------------------------
__global__ __launch_bounds__(TJ) void setconv_partial_kernel(
    const float* __restrict__ x_c, const float* __restrict__ y_c,
    const float* __restrict__ x_grid, const float* __restrict__ log_sigma,
    float* __restrict__ part)
{
    __shared__ __align__(16) float lds_x[ISLICE];
    __shared__ __align__(16) float lds_y[ISLICE * OUTD];
    __shared__ float lds_g[CH];

    const int t  = threadIdx.x;
    const int j  = blockIdx.x * TJ + t;
    const int s  = blockIdx.y;
    const int i0 = s * ISLICE;

    if (t < CH) {
        // inv_sigma2 = exp(-2*ls);  w = exp(-0.5*d*inv_sigma2) = exp2(d*gamma)
        float ls      = log_sigma[t];
        float inv_s2  = __builtin_amdgcn_exp2f(ls * (-2.0f * LOG2E));
        lds_g[t]      = -0.5f * LOG2E * inv_s2;
    }
    for (int k = t; k < ISLICE; k += TJ) lds_x[k] = x_c[i0 + k];
    for (int k = t; k < (ISLICE * OUTD) / 4; k += TJ)
        ((float4*)lds_y)[k] = ((const float4*)(y_c + (size_t)i0 * OUTD))[k];
    __syncthreads();

    float g[CH];
#pragma unroll
    for (int c = 0; c < CH; ++c) g[c] = lds_g[c];

    const float xg = x_grid[j];
    float acc[CH];
#pragma unroll
    for (int c = 0; c < CH; ++c) acc[c] = 0.0f;

    for (int i = 0; i < ISLICE; ++i) {
        // in_dim == 1: dist = sqrt(diff^2) = |diff|  (Laplacian kernel!)
        float d = fabsf(lds_x[i] - xg);
#pragma unroll
        for (int c = 0; c < CH; ++c) {
            float w  = __builtin_amdgcn_exp2f(d * g[c]);   // v_exp_f32
            float fc = (c == 0) ? 1.0f : lds_y[i * OUTD + (c - 1)];
            acc[c] += fc * w;
        }
    }

    float* p = part + ((size_t)s * NG + j) * KPAD;
#pragma unroll
    for (int c = 0; c < CH; ++c) p[c] = acc[c];
    p[33] = 0.0f; p[34] = 0.0f; p[35] = 0.0f;   // zero K-padding
}

// ---------------------------------------------------------------------------
// Kernel 2: reduce slices (fixed order => deterministic) + density-normalize.
//   F2[j] = [density, conv/(density+eps), 0,0,0]  (KPAD=36 wide)
// ---------------------------------------------------------------------------
__global__ __launch_bounds__(256) void reduce_norm_kernel(
    const float* __restrict__ part, float* __restrict__ F2)
{
    int j = blockIdx.x * blockDim.x + threadIdx.x;
    if (j >= NG) return;

    float sums[CH];
#pragma unroll
    for (int c = 0; c < CH; ++c) sums[c] = 0.0f;
    for (int s = 0; s < NSLICE; ++s) {
        const float* p = part + ((size_t)s * NG + j) * KPAD;
#pragma unroll
        for (int c = 0; c < CH; ++c) sums[c] += p[c];
    }

    float dens = sums[0];
    float inv  = 1.0f / (dens + 1e-8f);
    float* f   = F2 + (size_t)j * KPAD;
    f[0] = dens;
#pragma unroll
    for (int c = 1; c < CH; ++c) f[c] = sums[c] * inv;
    f[33] = 0.0f; f[34] = 0.0f; f[35] = 0.0f;
}

// ---------------------------------------------------------------------------
// Kernel 3: out = F2 @ W^T + b via V_WMMA_F32_16X16X4_F32 (full f32 precision).
// One wave per 16x16 output tile; 9 K-steps cover KPAD=36.
// Fragment layouts per CDNA5 ISA:
//   A (16x4 f32): v[0] = K=(0 | 2 for upper half-wave), v[1] = K=(1 | 3), M=lane&15
//   B (4x16 f32): mirrored K pattern, N=lane&15
//   C/D (16x16 f32, v8f): VGPR r -> M = r + 8*(lane>=16), N = lane&15
// Bias folded into the C accumulator init (each lane owns a fixed column).
// All loads are unconditional contiguous b64s (F2 and Wp are zero-padded).
// Block = 32 threads, zero divergence (WMMA requires EXEC all-1s).
// ---------------------------------------------------------------------------
__global__ __launch_bounds__(32) void wmma_proj_kernel(
    const float* __restrict__ F2, const float* __restrict__ Wp,
    const float* __restrict__ bias, float* __restrict__ out)
{
    const int lane = threadIdx.x;          // 0..31
    const int m    = lane & 15;
    const int half = lane >> 4;            // 0: lanes 0-15, 1: lanes 16-31
    const int j0   = blockIdx.x * 16;      // M tile (grid points)
    const int o0   = blockIdx.y * 16;      // N tile (output channels)

    const float* arow = F2 + (size_t)(j0 + m) * KPAD + 2 * half;
    const float* brow = Wp + (size_t)(o0 + m) * KPAD + 2 * half;

    // Fold bias into C: lane's column is fixed (= o0+m) across all 8 D VGPRs.
    const float bv = bias[o0 + m];
    v8f acc = {bv, bv, bv, bv, bv, bv, bv, bv};

#pragma unroll
    for (int kk = 0; kk < KPAD / 4; ++kk) {
        v2f a, bf;
        a.x  = arow[kk * 4 + 0];
        a.y  = arow[kk * 4 + 1];
        bf.x = brow[kk * 4 + 0];
        bf.y = brow[kk * 4 + 1];
        // 8 args: (neg_a, A, neg_b, B, c_mod, C, reuse_a, reuse_b)
        acc = __builtin_amdgcn_wmma_f32_16x16x4_f32(
            false, a, false, bf, (short)0, acc, false, false);
    }

#pragma unroll
    for (int r = 0; r < 8; ++r) {
        int row = j0 + r + 8 * half;
        out[(size_t)row * OUTD + (o0 + m)] = acc[r];
    }
}

// ---------------------------------------------------------------------------
extern "C" void kernel_launch(void* const* d_in, const int* in_sizes, int n_in,
                              void* d_out, int out_size, void* d_ws, size_t ws_size,
                              hipStream_t stream)
{
    const float* x_c       = (const float*)d_in[0];   // (1024,1)
    const float* y_c       = (const float*)d_in[1];   // (1024,32)
    const float* x_grid    = (const float*)d_in[2];   // (2048,1)
    const float* log_sigma = (const float*)d_in[3];   // (33,)
    const float* W         = (const float*)d_in[4];   // (32,33)
    const float* bias      = (const float*)d_in[5];   // (32,)
    float*       out       = (float*)d_out;           // (2048,32)

    float* part = (float*)d_ws;                                  // NSLICE*NG*KPAD
    float* F2   = part + (size_t)NSLICE * NG * KPAD;             // NG*KPAD
    float* Wp   = F2   + (size_t)NG * KPAD;                      // OUTD*KPAD
    // total ws use: (8*2048*36 + 2048*36 + 32*36)*4B ~= 2.7 MB

    prep_w_kernel<<<dim3((OUTD * KPAD + 255) / 256), 256, 0, stream>>>(W, Wp);
    setconv_partial_kernel<<<dim3(NG / TJ, NSLICE), TJ, 0, stream>>>(
        x_c, y_c, x_grid, log_sigma, part);
    reduce_norm_kernel<<<dim3(NG / 256), 256, 0, stream>>>(part, F2);
    wmma_proj_kernel<<<dim3(NG / 16, OUTD / 16), 32, 0, stream>>>(
        F2, Wp, bias, out);
}